// TERowParallelLinear_46222438039739
// MI455X (gfx1250) — compile-verified
//
#include <hip/hip_runtime.h>

// ---- CDNA5 (gfx1250) fp32 WMMA GEMM: out[M,N] = A[M,K] * W[N,K]^T + bias ----
// M = B*S = 8192, N = OUT = 4096, K = IN = 16384, all fp32.
// Memory-bound (~0.9 GiB mandatory traffic @ 23.3 TB/s), so full-precision
// V_WMMA_F32_16X16X4_F32 is the right matrix op (no reason to downcast).

typedef __attribute__((ext_vector_type(2))) float v2f;
typedef __attribute__((ext_vector_type(4))) float v4f;
typedef __attribute__((ext_vector_type(8))) float v8f;
typedef __attribute__((ext_vector_type(4))) int   v4i;

#if defined(__has_builtin)
#  if __has_builtin(__builtin_amdgcn_global_load_async_to_lds_b128)
#    define HAVE_ASYNC 1
#  else
#    define HAVE_ASYNC 0
#  endif
#else
#  define HAVE_ASYNC 0
#endif

#define M_TOT 8192
#define N_TOT 4096
#define K_TOT 16384

constexpr int BM  = 128;            // WG tile M
constexpr int BN  = 256;            // WG tile N
constexpr int BK  = 16;             // K slab per iteration (4 wmma k-steps)
constexpr int BKP = 20;             // padded LDS row (floats): spreads banks, keeps 8B align
constexpr int KITERS = K_TOT / BK;  // 1024
constexpr int MT = M_TOT / BM;      // 64
constexpr int NT = N_TOT / BN;      // 16
constexpr int GN = 8;               // N-tiles per L2 panel (8 * 16MiB weight slab < 192MB L2)

__device__ __forceinline__ void cp16_g2l(const float* g, float* l) {
#if HAVE_ASYNC
  // gfx1250 GLOBAL_LOAD_ASYNC_TO_LDS_B128 (tracked by ASYNCcnt)
  auto gp = (__attribute__((address_space(1))) v4i*)(unsigned long long)g;
  auto lp = (__attribute__((address_space(3))) v4i*)l;
  __builtin_amdgcn_global_load_async_to_lds_b128(gp, lp, 0, 0);
#else
  *(v4f*)l = *(const v4f*)g;   // global_load_b128 + ds_store_b128 fallback
#endif
}

__device__ __forceinline__ void wait_async0() {
#if HAVE_ASYNC
#  if __has_builtin(__builtin_amdgcn_s_wait_asynccnt)
  __builtin_amdgcn_s_wait_asynccnt(0);
#  else
  asm volatile("s_wait_asynccnt 0" ::: "memory");
#  endif
#endif
}

__global__ __launch_bounds__(256)
void rowlinear_wmma_f32(const float* __restrict__ A,
                        const float* __restrict__ W,
                        const float* __restrict__ bias,
                        float* __restrict__ C)
{
  __shared__ float sA[2][BM * BKP];   // 2 x 10 KB
  __shared__ float sW[2][BN * BKP];   // 2 x 20 KB

  const int tid  = threadIdx.x;
  const int lane = tid & 31;
  const int l16  = lane & 15;   // N (or M) index within 16
  const int lhi  = lane >> 4;   // lane-half selects K pair / M-half per ISA layout
  const int wave = tid >> 5;    // 8 waves
  const int wm   = (wave >> 2) * 64;  // wave M offset within WG tile (0,64)
  const int wn   = (wave & 3)  * 64;  // wave N offset within WG tile (0..192)

  // Panel swizzle for L2 reuse of the weight matrix:
  // consecutive blocks cover GN=8 N-tiles (128 MiB of W resident in L2)
  // while sweeping all 64 M-tiles => W read ~once, A read ~twice.
  const int bid       = blockIdx.x;
  const int panelSize = MT * GN;
  const int panel     = bid / panelSize;
  const int within    = bid - panel * panelSize;
  const int nt        = panel * GN + (within & (GN - 1));
  const int mt        = within / GN;

  const float* aTile = A + (size_t)mt * BM * K_TOT;
  const float* wTile = W + (size_t)nt * BN * K_TOT;

  auto loadTiles = [&](int buf, int kb) {
    const float* ab = aTile + kb * BK;
    const float* wb = wTile + kb * BK;
    // A tile: 128 rows x 16 floats = 512 float4 chunks, 2 per thread
#pragma unroll
    for (int i = 0; i < 2; ++i) {
      const int c = i * 256 + tid;
      const int r = c >> 2, q = c & 3;
      cp16_g2l(ab + (size_t)r * K_TOT + q * 4, &sA[buf][r * BKP + q * 4]);
    }
    // W tile: 256 rows x 16 floats = 1024 float4 chunks, 4 per thread
#pragma unroll
    for (int i = 0; i < 4; ++i) {
      const int c = i * 256 + tid;
      const int r = c >> 2, q = c & 3;
      cp16_g2l(wb + (size_t)r * K_TOT + q * 4, &sW[buf][r * BKP + q * 4]);
    }
  };

  const v8f vzero = {};
  v8f acc[4][4];
#pragma unroll
  for (int i = 0; i < 4; ++i)
#pragma unroll
    for (int j = 0; j < 4; ++j) acc[i][j] = vzero;

  loadTiles(0, 0);
  wait_async0();
  __syncthreads();

  for (int kb = 0; kb < KITERS; ++kb) {
    const int buf = kb & 1;
    if (kb + 1 < KITERS) loadTiles(buf ^ 1, kb + 1);

#pragma unroll
    for (int ks = 0; ks < 4; ++ks) {
      // ISA 32-bit A (16x4) layout: lane<16 -> K={k0,k0+1}, lane>=16 -> K={k0+2,k0+3}
      const int k0 = ks * 4 + lhi * 2;
      v2f a[4], b[4];
#pragma unroll
      for (int t = 0; t < 4; ++t)
        a[t] = *(const v2f*)&sA[buf][(wm + t * 16 + l16) * BKP + k0];
#pragma unroll
      for (int t = 0; t < 4; ++t)
        b[t] = *(const v2f*)&sW[buf][(wn + t * 16 + l16) * BKP + k0];

#pragma unroll
      for (int i = 0; i < 4; ++i)
#pragma unroll
        for (int j = 0; j < 4; ++j)
          acc[i][j] = __builtin_amdgcn_wmma_f32_16x16x4_f32(
              false, a[i], false, b[j], (short)0, acc[i][j], false, false);
    }

    if (kb + 1 < KITERS) {
      wait_async0();     // our async writes into next buffer have landed
      __syncthreads();   // everyone done computing current buffer + next buffer ready
    }
  }

  // Epilogue: C/D 16x16 f32 layout: lane<16 -> M=vgpr, lane>=16 -> M=vgpr+8; N=lane%16
  float bj[4];
#pragma unroll
  for (int j = 0; j < 4; ++j)
    bj[j] = bias[nt * BN + wn + j * 16 + l16];

  float* cBase = C + (size_t)(mt * BM + wm) * N_TOT + nt * BN + wn;
#pragma unroll
  for (int i = 0; i < 4; ++i) {
#pragma unroll
    for (int v = 0; v < 8; ++v) {
      const int row = i * 16 + lhi * 8 + v;
      float* cr = cBase + (size_t)row * N_TOT + l16;
#pragma unroll
      for (int j = 0; j < 4; ++j)
        cr[j * 16] = acc[i][j][v] + bj[j];
    }
  }
}

extern "C" void kernel_launch(void* const* d_in, const int* in_sizes, int n_in,
                              void* d_out, int out_size, void* d_ws, size_t ws_size,
                              hipStream_t stream) {
  const float* A    = (const float*)d_in[0];   // input_  [B,S,IN] = [8192,16384]
  const float* Wt   = (const float*)d_in[1];   // weight  [OUT,IN] = [4096,16384]
  const float* bias = (const float*)d_in[2];   // bias    [OUT]
  float*       out  = (float*)d_out;           // [8192,4096]

  dim3 grid(MT * NT);   // 64 * 16 = 1024 workgroups
  dim3 block(256);      // 8 waves (wave32)
  rowlinear_wmma_f32<<<grid, block, 0, stream>>>(A, Wt, bias, out);
}